// EADGNN_78469052498589
// MI455X (gfx1250) — compile-verified
//
#include <hip/hip_runtime.h>

typedef __attribute__((ext_vector_type(2))) float v2f;
typedef __attribute__((ext_vector_type(8))) float v8f;

// ---------------------------------------------------------------- norm setup

__global__ void init_deg_kernel(float* __restrict__ deg, int n) {
    int i = blockIdx.x * blockDim.x + threadIdx.x;
    if (i < n) deg[i] = 1.0f;   // self-loop contribution
}

__global__ void deg_accum_kernel(const int* __restrict__ col,
                                 float* __restrict__ deg, int e) {
    int i = blockIdx.x * blockDim.x + threadIdx.x;
    if (i < e) atomicAdd(&deg[col[i]], 1.0f);
}

__global__ void dinv_kernel(const float* __restrict__ deg,
                            float* __restrict__ dinv, int n) {
    int i = blockIdx.x * blockDim.x + threadIdx.x;
    if (i < n) dinv[i] = rsqrtf(deg[i]);   // deg >= 1 always (self loops)
}

__global__ void norm_kernel(const int* __restrict__ row,
                            const int* __restrict__ col,
                            const float* __restrict__ dinv,
                            float* __restrict__ nrm, int e) {
    int i = blockIdx.x * blockDim.x + threadIdx.x;
    if (i < e) nrm[i] = dinv[row[i]] * dinv[col[i]];
}

// ---------------------------------------------------------------- WMMA GEMM
// C[M,Nout] = A[M,K] @ W[K,Nout]; K multiple of 128, Nout multiple of 16.
// 128 threads = 4 waves; block tile 64 rows x 16 cols; one 16x16 tile / wave.

__global__ __launch_bounds__(128)
void gemm_wmma_kernel(const float* __restrict__ A, const float* __restrict__ W,
                      float* __restrict__ C, int M, int K, int Nout) {
    __shared__ float As[64][132];   // +4 pad: conflict-free frag reads
    __shared__ float Ws[128][16];

    const int tid  = threadIdx.x;
    const int lane = tid & 31;
    const int wave = tid >> 5;
    const int row0 = blockIdx.x * 64;
    const int col0 = blockIdx.y * 16;

    const int m  = lane & 15;            // M (or N) index within tile
    const int kk = (lane >> 4) << 1;     // K sub-offset: 0 (lanes 0-15) / 2 (16-31)

    v8f acc = {};

    for (int kc = 0; kc < K; kc += 128) {
        // ---- stage A chunk: 64 x 128 floats = 2048 float4, coalesced
        #pragma unroll
        for (int it = 0; it < 16; ++it) {
            int idx = tid + it * 128;        // 0..2047
            int r   = idx >> 5;              // 32 float4 per row
            int c4  = idx & 31;
            float4 v = make_float4(0.f, 0.f, 0.f, 0.f);
            int grow = row0 + r;
            if (grow < M)
                v = *(const float4*)(A + (size_t)grow * K + kc + c4 * 4);
            As[r][c4 * 4 + 0] = v.x; As[r][c4 * 4 + 1] = v.y;
            As[r][c4 * 4 + 2] = v.z; As[r][c4 * 4 + 3] = v.w;
        }
        // ---- stage W chunk: 128 x 16 floats = 512 float4
        #pragma unroll
        for (int it = 0; it < 4; ++it) {
            int idx = tid + it * 128;        // 0..511
            int kr  = idx >> 2;
            int c4  = idx & 3;
            float4 v = *(const float4*)(W + (size_t)(kc + kr) * Nout + col0 + c4 * 4);
            Ws[kr][c4 * 4 + 0] = v.x; Ws[kr][c4 * 4 + 1] = v.y;
            Ws[kr][c4 * 4 + 2] = v.z; Ws[kr][c4 * 4 + 3] = v.w;
        }
        __syncthreads();

        const int mrow = wave * 16 + m;
        #pragma unroll
        for (int k0 = 0; k0 < 128; k0 += 4) {
            v2f a, b;
            // A 16x4 fp32 frag: lanes 0-15 K=k0,k0+1 ; lanes 16-31 K=k0+2,k0+3
            a.x = As[mrow][k0 + kk];
            a.y = As[mrow][k0 + kk + 1];
            // B 4x16 fp32 frag (mirrored layout): lane n holds column n
            b.x = Ws[k0 + kk][m];
            b.y = Ws[k0 + kk + 1][m];
            acc = __builtin_amdgcn_wmma_f32_16x16x4_f32(
                false, a, false, b, (short)0, acc, false, false);
        }
        __syncthreads();
    }

    // C frag: VGPR r -> lanes 0-15: M=r ; lanes 16-31: M=r+8
    const int nc    = lane & 15;
    const int mhalf = (lane >> 4) * 8;
    #pragma unroll
    for (int r = 0; r < 8; ++r) {
        int grow = row0 + wave * 16 + mhalf + r;
        if (grow < M)
            C[(size_t)grow * Nout + col0 + nc] = acc[r];
    }
}

// ---------------------------------------------------------------- SpMM parts

// out[i][:] = dinv[i]^2 * h[i][:]   (self-loop term; also zero-initializes)
__global__ void self_init_kernel(const float* __restrict__ h,
                                 const float* __restrict__ dinv,
                                 float* __restrict__ out,
                                 long long total, int f, int shift) {
    long long i = (long long)blockIdx.x * blockDim.x + threadIdx.x;
    if (i >= total) return;
    int node = (int)(i >> shift);
    int c4   = (int)(i & ((1 << shift) - 1));
    float s = dinv[node]; s *= s;
    float4 v = *(const float4*)(h + (size_t)node * f + c4 * 4);
    float4 o; o.x = v.x * s; o.y = v.y * s; o.z = v.z * s; o.w = v.w * s;
    *(float4*)(out + (size_t)node * f + c4 * 4) = o;
}

// out[col[e]][:] += norm[e] * h[row[e]][:]
__global__ void scatter_kernel(const float* __restrict__ h,
                               const int* __restrict__ row,
                               const int* __restrict__ col,
                               const float* __restrict__ nrm,
                               float* __restrict__ out,
                               long long total, int f, int shift) {
    long long i = (long long)blockIdx.x * blockDim.x + threadIdx.x;
    if (i >= total) return;
    int e  = (int)(i >> shift);
    int c4 = (int)(i & ((1 << shift) - 1));
    float w = nrm[e];
    int rs = row[e], cs = col[e];
    float4 v = *(const float4*)(h + (size_t)rs * f + c4 * 4);
    float* o = out + (size_t)cs * f + c4 * 4;
    atomicAdd(o + 0, v.x * w);
    atomicAdd(o + 1, v.y * w);
    atomicAdd(o + 2, v.z * w);
    atomicAdd(o + 3, v.w * w);
}

__global__ void bias_act_kernel(float* __restrict__ x,
                                const float* __restrict__ bias,
                                int total, int fmask, int do_relu) {
    int i = blockIdx.x * blockDim.x + threadIdx.x;
    if (i >= total) return;
    float v = x[i] + bias[i & fmask];
    x[i] = do_relu ? fmaxf(v, 0.0f) : v;
}

// ---------------------------------------------------------------- launch

extern "C" void kernel_launch(void* const* d_in, const int* in_sizes, int n_in,
                              void* d_out, int out_size, void* d_ws, size_t ws_size,
                              hipStream_t stream) {
    (void)n_in; (void)out_size; (void)ws_size;

    const int*   edge_index = (const int*)d_in[0];
    const float* x_in       = (const float*)d_in[1];
    const float* W1         = (const float*)d_in[2];
    const float* b1         = (const float*)d_in[3];
    const float* W2         = (const float*)d_in[4];
    const float* b2         = (const float*)d_in[5];

    const int D = 128, F = 512;
    const int E = in_sizes[0] / 2;
    const int N = in_sizes[1] / D;
    const int L = in_sizes[2] / (D * F);

    const int* row = edge_index;        // sources
    const int* col = edge_index + E;    // targets

    float* ws   = (float*)d_ws;
    float* deg  = ws;                ws += N;
    float* dinv = ws;                ws += N;
    float* nrm  = ws;                ws += E;
    float* h    = ws;                ws += (size_t)N * F;   // GEMM1 out / GEMM2 out
    float* acc  = ws;                ws += (size_t)N * F;   // scatter accumulator
    float* xbuf = ws;                                        // N*D inter-layer x

    const int T = 256;

    // --- gcn_norm ---
    init_deg_kernel<<<(N + T - 1) / T, T, 0, stream>>>(deg, N);
    deg_accum_kernel<<<(E + T - 1) / T, T, 0, stream>>>(col, deg, E);
    dinv_kernel<<<(N + T - 1) / T, T, 0, stream>>>(deg, dinv, N);
    norm_kernel<<<(E + T - 1) / T, T, 0, stream>>>(row, col, dinv, nrm, E);

    const float* xcur = x_in;
    for (int l = 0; l < L; ++l) {
        // h = xcur @ W1[l]   [N,128] x [128,512]
        dim3 g1((N + 63) / 64, F / 16);
        gemm_wmma_kernel<<<g1, 128, 0, stream>>>(
            xcur, W1 + (size_t)l * D * F, h, N, D, F);

        // acc = Â h  (self + edges), then relu(acc + b1)
        long long tn = (long long)N * (F / 4);
        self_init_kernel<<<(int)((tn + T - 1) / T), T, 0, stream>>>(
            h, dinv, acc, tn, F, 7);
        long long te = (long long)E * (F / 4);
        scatter_kernel<<<(int)((te + T - 1) / T), T, 0, stream>>>(
            h, row, col, nrm, acc, te, F, 7);
        bias_act_kernel<<<(N * F + T - 1) / T, T, 0, stream>>>(
            acc, b1 + (size_t)l * F, N * F, F - 1, 1);

        // g2 = acc @ W2[l]   [N,512] x [512,128] -> reuse h region
        float* g2 = h;
        dim3 g2d((N + 63) / 64, D / 16);
        gemm_wmma_kernel<<<g2d, 128, 0, stream>>>(
            acc, W2 + (size_t)l * F * D, g2, N, F, D);

        // xout = Â g2 + b2
        float* xout = (l == L - 1) ? (float*)d_out : xbuf;
        long long tn2 = (long long)N * (D / 4);
        self_init_kernel<<<(int)((tn2 + T - 1) / T), T, 0, stream>>>(
            g2, dinv, xout, tn2, D, 5);
        long long te2 = (long long)E * (D / 4);
        scatter_kernel<<<(int)((te2 + T - 1) / T), T, 0, stream>>>(
            g2, row, col, nrm, xout, te2, D, 5);
        bias_act_kernel<<<(N * D + T - 1) / T, T, 0, stream>>>(
            xout, b2 + (size_t)l * D, N * D, D - 1, 0);

        xcur = xout;
    }
}